// dynamic_balance_focal_loss_1992864825406
// MI455X (gfx1250) — compile-verified
//
#include <hip/hip_runtime.h>
#include <hip/hip_bf16.h>
#include <cstdint>
#include <cstddef>

// ---------------- problem constants ----------------
#define BATCH        262144
#define NCLS         1000
#define NVEC         250      // float4 chunks per row (250*16B = 4000B, 16B aligned per row)
#define WAVES_PER_BLOCK 4
#define ROWS_PER_WAVE   8
#define BLOCK        (WAVES_PER_BLOCK * 32)          // 128 threads
#define ROWS_PER_BLOCK (WAVES_PER_BLOCK * ROWS_PER_WAVE) // 32 rows
#define GRID         (BATCH / ROWS_PER_BLOCK)        // 8192 blocks

// K=0.5, T=3.0, ALPHA=1.0, GAMMA=2.0 from the reference.

// Issue an async DMA copy of one 1000-float row (250 x B128) from global to LDS.
// CDNA5 path: GLOBAL_LOAD_ASYNC_TO_LDS_B128, tracked by ASYNCcnt.
// th:TH_LOAD_NT — the logits stream is read exactly once; keep it from evicting
// the hot weight/mask tables (5 MB) out of the 192 MB L2.
__device__ __forceinline__ void async_copy_row(const float* __restrict__ gbase,
                                               unsigned ldsBase, int lane) {
#pragma unroll
  for (int i = 0; i < 8; ++i) {
    int vec = i * 32 + lane;                 // float4 index within row
    if (vec < NVEC) {
      unsigned long long ga = (unsigned long long)(uintptr_t)(gbase + vec * 4);
      unsigned la = ldsBase + (unsigned)(vec * 16);
      asm volatile("global_load_async_to_lds_b128 %0, %1, off th:TH_LOAD_NT"
                   :: "v"(la), "v"(ga) : "memory");
    }
  }
}

__global__ void __launch_bounds__(BLOCK)
focal_main_kernel(const float* __restrict__ logits,      // [BATCH, NCLS]
                  const int* __restrict__ target,        // [BATCH]
                  const float* __restrict__ wtab,        // [NCLS, NCLS]
                  const unsigned char* __restrict__ mtab,// [NCLS, NCLS] bool (1 byte)
                  float* __restrict__ partials)          // [GRID*2]
{
  __shared__ __align__(16) float smem[WAVES_PER_BLOCK][2][1024]; // 32 KB double buffer
  __shared__ float red1[WAVES_PER_BLOCK];
  __shared__ float red2[WAVES_PER_BLOCK];

  const int wave = threadIdx.x >> 5;
  const int lane = threadIdx.x & 31;
  const int rowBase = blockIdx.x * ROWS_PER_BLOCK + wave * ROWS_PER_WAVE;

  // Low 32 bits of a generic LDS pointer == wave-relative LDS byte address
  // (same convention DS ops and the async-LDS VDST operand use).
  const unsigned ldsBase0 = (unsigned)(uintptr_t)&smem[wave][0][0];
  const unsigned ldsBase1 = (unsigned)(uintptr_t)&smem[wave][1][0];

  float accL1 = 0.f, accL2 = 0.f;

  // Prologue: DMA row 0 into buffer 0.
  async_copy_row(logits + (size_t)rowBase * NCLS, ldsBase0, lane);

  for (int j = 0; j < ROWS_PER_WAVE; ++j) {
    const int row = rowBase + j;

    // Prefetch next row into the alternate buffer, then wait for current row.
    if (j + 1 < ROWS_PER_WAVE) {
      async_copy_row(logits + (size_t)(row + 1) * NCLS,
                     (((j + 1) & 1) ? ldsBase1 : ldsBase0), lane);
      asm volatile("s_wait_asynccnt 8" ::: "memory"); // next row's 8 DMAs may stay in flight
    } else {
      asm volatile("s_wait_asynccnt 0" ::: "memory");
    }

    const float* buf = &smem[wave][j & 1][0];

    // ---- pass 1: row max + first argmax (from LDS, B128 reads) ----
    float vmax = -3.402823466e38f;
    int   vidx = 0x7fffffff;
#pragma unroll
    for (int i = 0; i < 8; ++i) {
      int vec = i * 32 + lane;
      if (vec < NVEC) {
        float4 x = *(const float4*)(buf + vec * 4);
        int b0 = vec * 4;
        if (x.x > vmax) { vmax = x.x; vidx = b0;     }
        if (x.y > vmax) { vmax = x.y; vidx = b0 + 1; }
        if (x.z > vmax) { vmax = x.z; vidx = b0 + 2; }
        if (x.w > vmax) { vmax = x.w; vidx = b0 + 3; }
      }
    }
#pragma unroll
    for (int o = 16; o > 0; o >>= 1) {  // wave32 xor-shuffle reduce
      float om = __shfl_xor(vmax, o, 32);
      int   oi = __shfl_xor(vidx, o, 32);
      if (om > vmax || (om == vmax && oi < vidx)) { vmax = om; vidx = oi; }
    }

    // ---- pass 2: sum exp(x - max) ----
    float s = 0.f;
#pragma unroll
    for (int i = 0; i < 8; ++i) {
      int vec = i * 32 + lane;
      if (vec < NVEC) {
        float4 x = *(const float4*)(buf + vec * 4);
        s += __expf(x.x - vmax) + __expf(x.y - vmax)
           + __expf(x.z - vmax) + __expf(x.w - vmax);
      }
    }
#pragma unroll
    for (int o = 16; o > 0; o >>= 1) s += __shfl_xor(s, o, 32);

    // ---- per-row epilogue (lane 0) ----
    if (lane == 0) {
      int   tgt = target[row];
      float xt  = buf[tgt];                   // target logit straight from LDS (no global re-read)
      float ce  = vmax + __logf(s) - xt;      // -log_softmax[target]
      float pt  = __expf(-ce);
      float om  = 1.f - pt;
      float f   = om * om * ce;               // ALPHA=1, GAMMA=2
      int   ix  = tgt * NCLS + vidx;          // [target, pred]
      float w   = wtab[ix];
      bool  m   = (mtab[ix] != 0);
      accL1 += m ? f * w : 0.f;
      accL2 += m ? 0.f  : f;
    }
  }

  if (lane == 0) { red1[wave] = accL1; red2[wave] = accL2; }
  __syncthreads();
  if (threadIdx.x == 0) {
    float a = 0.f, b = 0.f;
#pragma unroll
    for (int w = 0; w < WAVES_PER_BLOCK; ++w) { a += red1[w]; b += red2[w]; }
    partials[2 * blockIdx.x]     = a;   // deterministic: unique slot per block
    partials[2 * blockIdx.x + 1] = b;
  }
}

__global__ void __launch_bounds__(256)
focal_finalize_kernel(const float* __restrict__ partials, float* __restrict__ out)
{
  __shared__ float s1[256];
  __shared__ float s2[256];
  float a = 0.f, b = 0.f;
  for (int i = threadIdx.x; i < GRID; i += 256) {
    a += partials[2 * i];
    b += partials[2 * i + 1];
  }
  s1[threadIdx.x] = a; s2[threadIdx.x] = b;
  __syncthreads();
  for (int o = 128; o > 0; o >>= 1) {
    if ((int)threadIdx.x < o) {
      s1[threadIdx.x] += s1[threadIdx.x + o];
      s2[threadIdx.x] += s2[threadIdx.x + o];
    }
    __syncthreads();
  }
  if (threadIdx.x == 0) {
    float loss1 = s1[0], loss2 = s2[0];
    const float invB = 1.f / (float)BATCH;
    float l1 = loss1 * invB;
    float l2 = loss2 * invB;
    float w2 = 1.f / (1.f + __expf(-(0.5f * (l1 - 3.0f))));  // sigmoid(K*(l1-T))
    out[0] = (loss1 > 0.f) ? (l1 + w2 * l2) : (loss2 * invB);
  }
}

extern "C" void kernel_launch(void* const* d_in, const int* in_sizes, int n_in,
                              void* d_out, int out_size, void* d_ws, size_t ws_size,
                              hipStream_t stream) {
  (void)in_sizes; (void)n_in; (void)out_size; (void)ws_size;
  const float*         logits = (const float*)d_in[0];         // [262144,1000] f32
  const int*           target = (const int*)d_in[1];           // [262144] i32
  const float*         wtab   = (const float*)d_in[2];         // [1000,1000] f32
  const unsigned char* mtab   = (const unsigned char*)d_in[3]; // [1000,1000] bool
  float* out      = (float*)d_out;
  float* partials = (float*)d_ws;                              // GRID*2 floats = 64 KB

  focal_main_kernel<<<GRID, BLOCK, 0, stream>>>(logits, target, wtab, mtab, partials);
  focal_finalize_kernel<<<1, 256, 0, stream>>>(partials, out);
}